// spline_net_30210799960814
// MI455X (gfx1250) — compile-verified
//
#include <hip/hip_runtime.h>
#include <hip/hip_bf16.h>

typedef __attribute__((ext_vector_type(2))) float v2f;
typedef __attribute__((ext_vector_type(8))) float v8f;

// ---------------------------------------------------------------------------
// Layer-1 GEMMs: H0 = x@W1[0], H1 = x@W1[1], R1 = x@root1 + b1
// x: N x 128 (f32), weights 128 x 16. Full-precision fp32 WMMA (16x16x4).
// One 16-row tile per wave, 4 waves per block. Weights staged in LDS.
// ---------------------------------------------------------------------------
__global__ __launch_bounds__(128) void spline_gemm1_kernel(
    const float* __restrict__ x,
    const float* __restrict__ W1,     // 2 x 128 x 16
    const float* __restrict__ root1,  // 128 x 16
    const float* __restrict__ b1,     // 16
    float* __restrict__ H0, float* __restrict__ H1, float* __restrict__ R1,
    int N)
{
    __shared__ float w0s[128 * 16];
    __shared__ float w1s[128 * 16];
    __shared__ float wrs[128 * 16];

    const int t = threadIdx.x;
    for (int i = t; i < 2048; i += 128) {
        w0s[i] = W1[i];
        w1s[i] = W1[2048 + i];
        wrs[i] = root1[i];
    }
    __syncthreads();

    const int wave = t >> 5;
    const int lane = t & 31;
    const int tile = blockIdx.x * 4 + wave;
    const int row0 = tile * 16;
    if (row0 >= N) return;              // uniform per-wave: EXEC stays all-1s

    const int m    = lane & 15;         // M row within tile (A layout)
    const int koff = (lane >> 4) * 2;   // lanes 16-31 hold K=2,3 of each group
    const int n    = lane & 15;         // N column (B/C layout)

    const float* __restrict__ xrow = x + (size_t)(row0 + m) * 128;

    v8f c0 = {}; v8f c1 = {}; v8f cr = {};

#pragma unroll 4
    for (int k = 0; k < 128; k += 4) {
        v2f a;                              // A 16x4 f32 fragment (2 VGPRs)
        a.x = xrow[k + koff];
        a.y = xrow[k + koff + 1];

        const int kr = (k + koff) * 16 + n; // B 4x16 f32 fragment from LDS
        v2f b0, b1v, br;
        b0.x  = w0s[kr]; b0.y  = w0s[kr + 16];
        b1v.x = w1s[kr]; b1v.y = w1s[kr + 16];
        br.x  = wrs[kr]; br.y  = wrs[kr + 16];

        c0 = __builtin_amdgcn_wmma_f32_16x16x4_f32(false, a, false, b0,  (short)0, c0, false, false);
        c1 = __builtin_amdgcn_wmma_f32_16x16x4_f32(false, a, false, b1v, (short)0, c1, false, false);
        cr = __builtin_amdgcn_wmma_f32_16x16x4_f32(false, a, false, br,  (short)0, cr, false, false);
    }

    const float bias = b1[n];
#pragma unroll
    for (int r = 0; r < 8; ++r) {
        const int mm = r + 8 * (lane >> 4);   // C/D layout: VGPR r -> rows r, r+8
        const size_t o = (size_t)(row0 + mm) * 16 + n;
        H0[o] = c0[r];
        H1[o] = c1[r];
        R1[o] = cr[r] + bias;
    }
}

// ---------------------------------------------------------------------------
// Layer-2 GEMMs: h (N x 16) @ 16 x 10 weights, zero-padded to 16 cols in LDS.
// ---------------------------------------------------------------------------
__global__ __launch_bounds__(128) void spline_gemm2_kernel(
    const float* __restrict__ h,
    const float* __restrict__ W2,     // 2 x 16 x 10
    const float* __restrict__ root2,  // 16 x 10
    const float* __restrict__ b2,     // 10
    float* __restrict__ G0, float* __restrict__ G1, float* __restrict__ R2,
    int N)
{
    __shared__ float w0s[16 * 16];
    __shared__ float w1s[16 * 16];
    __shared__ float wrs[16 * 16];

    const int t = threadIdx.x;
    for (int i = t; i < 256; i += 128) {
        const int kk = i >> 4, nn = i & 15;
        float v0 = 0.f, v1 = 0.f, vr = 0.f;
        if (nn < 10) {
            v0 = W2[kk * 10 + nn];
            v1 = W2[160 + kk * 10 + nn];
            vr = root2[kk * 10 + nn];
        }
        w0s[i] = v0; w1s[i] = v1; wrs[i] = vr;
    }
    __syncthreads();

    const int wave = t >> 5;
    const int lane = t & 31;
    const int tile = blockIdx.x * 4 + wave;
    const int row0 = tile * 16;
    if (row0 >= N) return;

    const int m    = lane & 15;
    const int koff = (lane >> 4) * 2;
    const int n    = lane & 15;

    const float* __restrict__ hrow = h + (size_t)(row0 + m) * 16;

    v8f c0 = {}; v8f c1 = {}; v8f cr = {};

#pragma unroll
    for (int k = 0; k < 16; k += 4) {
        v2f a;
        a.x = hrow[k + koff];
        a.y = hrow[k + koff + 1];

        const int kr = (k + koff) * 16 + n;
        v2f b0, b1v, br;
        b0.x  = w0s[kr]; b0.y  = w0s[kr + 16];
        b1v.x = w1s[kr]; b1v.y = w1s[kr + 16];
        br.x  = wrs[kr]; br.y  = wrs[kr + 16];

        c0 = __builtin_amdgcn_wmma_f32_16x16x4_f32(false, a, false, b0,  (short)0, c0, false, false);
        c1 = __builtin_amdgcn_wmma_f32_16x16x4_f32(false, a, false, b1v, (short)0, c1, false, false);
        cr = __builtin_amdgcn_wmma_f32_16x16x4_f32(false, a, false, br,  (short)0, cr, false, false);
    }

    if (n < 10) {
        const float bias = b2[n];
#pragma unroll
        for (int r = 0; r < 8; ++r) {
            const int mm = r + 8 * (lane >> 4);
            const size_t o = (size_t)(row0 + mm) * 10 + n;
            G0[o] = c0[r];
            G1[o] = c1[r];
            R2[o] = cr[r] + bias;
        }
    }
}

// ---------------------------------------------------------------------------
// Edge gather + lerp + atomic scatter-sum. 16 lanes cooperate per edge.
//   msg = (1-u)*Ha[src] + u*Hb[src];  AGG[dst] += msg;  DEG[dst] += 1 (lane 0)
// ---------------------------------------------------------------------------
__global__ __launch_bounds__(256) void spline_edge_kernel(
    const float* __restrict__ Ha, const float* __restrict__ Hb,
    const long long* __restrict__ src, const long long* __restrict__ dst,
    const float* __restrict__ u,
    float* __restrict__ AGG, float* __restrict__ DEG,
    int E, int F)
{
    const int idx = blockIdx.x * 256 + threadIdx.x;
    const int e = idx >> 4;
    const int j = idx & 15;
    if (e >= E) return;

    const long long s = src[e];
    const long long d = dst[e];
    const float w1 = u[e];
    const float w0 = 1.0f - w1;

    if (j < F) {
        const float msg = w0 * Ha[(size_t)s * F + j] + w1 * Hb[(size_t)s * F + j];
        atomicAdd(&AGG[(size_t)d * F + j], msg);
    }
    if (DEG != nullptr && j == 0) {
        atomicAdd(&DEG[d], 1.0f);
    }
}

// ---------------------------------------------------------------------------
// Finalize layer 1: h = elu(AGG/max(DEG,1) + R1)
// ---------------------------------------------------------------------------
__global__ __launch_bounds__(256) void spline_fin1_kernel(
    const float* __restrict__ AGG, const float* __restrict__ DEG,
    const float* __restrict__ R1, float* __restrict__ h, int N)
{
    const int idx = blockIdx.x * 256 + threadIdx.x;
    if (idx >= N * 16) return;
    const int node = idx >> 4;
    float deg = DEG[node];
    deg = deg > 1.0f ? deg : 1.0f;
    const float v = AGG[idx] / deg + R1[idx];
    h[idx] = v > 0.0f ? v : (__expf(v) - 1.0f);
}

// ---------------------------------------------------------------------------
// Finalize layer 2: out = AGG/max(DEG,1) + R2   (F = 10)
// ---------------------------------------------------------------------------
__global__ __launch_bounds__(256) void spline_fin2_kernel(
    const float* __restrict__ AGG, const float* __restrict__ DEG,
    const float* __restrict__ R2, float* __restrict__ out, int N)
{
    const int idx = blockIdx.x * 256 + threadIdx.x;
    if (idx >= N * 10) return;
    const int node = idx / 10;
    float deg = DEG[node];
    deg = deg > 1.0f ? deg : 1.0f;
    out[idx] = AGG[idx] / deg + R2[idx];
}

// ---------------------------------------------------------------------------
// Launch
// inputs: x, edge_index(int64), edge_attr, W1, root1, b1, W2, root2, b2
// ---------------------------------------------------------------------------
extern "C" void kernel_launch(void* const* d_in, const int* in_sizes, int n_in,
                              void* d_out, int out_size, void* d_ws, size_t ws_size,
                              hipStream_t stream) {
    const float*     x     = (const float*)d_in[0];
    const long long* eidx  = (const long long*)d_in[1];   // int64 per reference
    const float*     u     = (const float*)d_in[2];
    const float*     W1    = (const float*)d_in[3];
    const float*     root1 = (const float*)d_in[4];
    const float*     b1    = (const float*)d_in[5];
    const float*     W2    = (const float*)d_in[6];
    const float*     root2 = (const float*)d_in[7];
    const float*     b2    = (const float*)d_in[8];
    float*           out   = (float*)d_out;

    const int N = in_sizes[0] / 128;   // 100000
    const int E = in_sizes[1] / 2;     // 1600000

    const long long* src = eidx;
    const long long* dst = eidx + E;

    // Workspace layout (floats)
    float* ws = (float*)d_ws;
    size_t o = 0;
    float* H0   = ws + o; o += (size_t)N * 16;
    float* H1   = ws + o; o += (size_t)N * 16;
    float* R1   = ws + o; o += (size_t)N * 16;
    float* AGG1 = ws + o; o += (size_t)N * 16;
    float* HACT = ws + o; o += (size_t)N * 16;
    float* DEG  = ws + o; o += (size_t)N;
    float* G0   = ws + o; o += (size_t)N * 10;
    float* G1   = ws + o; o += (size_t)N * 10;
    float* R2   = ws + o; o += (size_t)N * 10;
    float* AGG2 = ws + o; o += (size_t)N * 10;
    (void)ws_size; (void)n_in; (void)out_size;

    // Zero accumulators (async memset is graph-capture legal)
    hipMemsetAsync(AGG1, 0, (size_t)N * 16 * sizeof(float), stream);
    hipMemsetAsync(DEG,  0, (size_t)N * sizeof(float), stream);
    hipMemsetAsync(AGG2, 0, (size_t)N * 10 * sizeof(float), stream);

    const int tiles   = (N + 15) / 16;
    const int gblocks = (tiles + 3) / 4;

    // Layer 1
    spline_gemm1_kernel<<<gblocks, 128, 0, stream>>>(x, W1, root1, b1, H0, H1, R1, N);
    {
        const long long thr = (long long)E * 16;
        const int blocks = (int)((thr + 255) / 256);
        spline_edge_kernel<<<blocks, 256, 0, stream>>>(H0, H1, src, dst, u, AGG1, DEG, E, 16);
    }
    spline_fin1_kernel<<<(N * 16 + 255) / 256, 256, 0, stream>>>(AGG1, DEG, R1, HACT, N);

    // Layer 2
    spline_gemm2_kernel<<<gblocks, 128, 0, stream>>>(HACT, W2, root2, b2, G0, G1, R2, N);
    {
        const long long thr = (long long)E * 16;
        const int blocks = (int)((thr + 255) / 256);
        spline_edge_kernel<<<blocks, 256, 0, stream>>>(G0, G1, src, dst, u, AGG2, nullptr, E, 10);
    }
    spline_fin2_kernel<<<(N * 10 + 255) / 256, 256, 0, stream>>>(AGG2, DEG, R2, out, N);
}